// Maskedwords_13950053778295
// MI455X (gfx1250) — compile-verified
//
#include <hip/hip_runtime.h>

// Masked word dropout: out[i] = mask[i] ? 22 : x[i], over 33.55M int32 elems.
// Memory-bound streaming kernel: ~302 MB of traffic, ~13 us floor at 23.3 TB/s.
// Strategy: 4 elements/thread via b128 loads/stores, NT temporal hints (data is
// touched exactly once), global_prefetch ahead, wave32-friendly 256-thread blocks.

typedef int    i4v __attribute__((ext_vector_type(4)));
typedef float  f4v __attribute__((ext_vector_type(4)));

__global__ void __launch_bounds__(256)
Maskedwords_13950053778295_kernel(const int* __restrict__ x,
                                  const unsigned char* __restrict__ mask,
                                  float* __restrict__ out,
                                  int n4, int n) {
    const int i4 = blockIdx.x * blockDim.x + threadIdx.x;

    if (i4 < n4) {
        // Prefetch ~8KB ahead on the heavy stream (gfx1250 global_prefetch_b8).
        __builtin_prefetch(x + 4 * i4 + 2048, /*rw=*/0, /*locality=*/0);

        // 128-bit non-temporal load of 4 int32 words.
        const i4v xv = __builtin_nontemporal_load((const i4v*)x + i4);
        // 4 mask bytes as one 32-bit word (jax/numpy bool = 1 byte/elem).
        const unsigned int m4 =
            __builtin_nontemporal_load((const unsigned int*)mask + i4);

        f4v o;
        o.x = (m4 & 0x000000FFu) ? 22.0f : (float)xv.x;
        o.y = (m4 & 0x0000FF00u) ? 22.0f : (float)xv.y;
        o.z = (m4 & 0x00FF0000u) ? 22.0f : (float)xv.z;
        o.w = (m4 & 0xFF000000u) ? 22.0f : (float)xv.w;

        // 128-bit non-temporal store.
        __builtin_nontemporal_store(o, (f4v*)out + i4);
    }

    // Scalar tail (n not divisible by 4; empty for the 4096x8192 case).
    const int ti = n4 * 4 + i4;
    if (ti < n) {
        out[ti] = mask[ti] ? 22.0f : (float)x[ti];
    }
}

extern "C" void kernel_launch(void* const* d_in, const int* in_sizes, int n_in,
                              void* d_out, int out_size, void* d_ws, size_t ws_size,
                              hipStream_t stream) {
    const int*           x    = (const int*)d_in[0];
    const unsigned char* mask = (const unsigned char*)d_in[1];
    float*               out  = (float*)d_out;

    const int n  = in_sizes[0];      // 4096 * 8192 = 33,554,432
    const int n4 = n >> 2;           // 4 elements per thread

    const int threads = 256;         // 8 wave32 waves per block
    int blocks = (n4 + threads - 1) / threads;
    if (blocks < 1) blocks = 1;

    Maskedwords_13950053778295_kernel<<<blocks, threads, 0, stream>>>(
        x, mask, out, n4, n);
}